// SSD_60662118089051
// MI455X (gfx1250) — compile-verified
//
#include <hip/hip_runtime.h>
#include <hip/hip_bf16.h>
#include <math.h>

// ---------------------------------------------------------------------------
// SSD post-processing for MI455X (gfx1250, wave32).
//  K1: fused softmax(21) + box decode, scores written class-major (B,20,N)
//  K2: per-(b,class) exact top-200 via 4-pass MSD radix select + bitonic sort
//  K3: per-(b,class) NMS: 200x200 suppression bitmasks built with
//      V_WMMA_F32_16X16X4_F32 outer-sum tiles (union = D - inter) + wave32
//      ballot packing; then a cheap 200-step serial bit-scan.
//      IoU threshold test done divide-free: inter > t * union.
// ---------------------------------------------------------------------------

typedef __attribute__((ext_vector_type(2))) float v2f;
typedef __attribute__((ext_vector_type(8))) float v8f;

#define NUM_CLASSES 21
#define NCM1        20
#define K_TOP       200
#define KPAD        208   // 13 tiles of 16
#define NWORDS      7     // 224 bits >= 208 columns
#define CONF_TH     0.01f
#define NMS_TH      0.45f

static __device__ __forceinline__ unsigned float_key(float f) {
  unsigned u = __float_as_uint(f);
  return (u & 0x80000000u) ? ~u : (u | 0x80000000u);  // monotone: larger f -> larger key
}

static __device__ __forceinline__ unsigned ballot32(bool p) {
#if defined(__has_builtin)
#if __has_builtin(__builtin_amdgcn_ballot_w32)
  return __builtin_amdgcn_ballot_w32(p);
#else
  return (unsigned)__ballot(p);
#endif
#else
  return (unsigned)__ballot(p);
#endif
}

// ---------------------------------------------------------------------------
// Kernel 1: softmax over 21 classes + box decode.
// ---------------------------------------------------------------------------
__global__ void k_softmax_decode(const float* __restrict__ loc,
                                 const float* __restrict__ conf,
                                 const float* __restrict__ priors,
                                 float* __restrict__ scores_t,   // (B,20,N)
                                 float* __restrict__ boxes,      // (B,N,4)
                                 int B, int N) {
  int gid = blockIdx.x * blockDim.x + threadIdx.x;
  if (gid >= B * N) return;
  int b = gid / N;
  int n = gid - b * N;

  const float* c = conf + (size_t)gid * NUM_CLASSES;
  float v[NUM_CLASSES];
  float m = -INFINITY;
#pragma unroll
  for (int i = 0; i < NUM_CLASSES; ++i) { v[i] = c[i]; m = fmaxf(m, v[i]); }
  float s = 0.f;
#pragma unroll
  for (int i = 0; i < NUM_CLASSES; ++i) { v[i] = __expf(v[i] - m); s += v[i]; }
  float inv = 1.0f / s;

  size_t base = (size_t)b * NCM1 * N + n;   // class-major so K2 reads coalesced
#pragma unroll
  for (int i = 1; i < NUM_CLASSES; ++i)
    scores_t[base + (size_t)(i - 1) * N] = v[i] * inv;

  const float* L = loc + (size_t)gid * 4;
  const float* P = priors + (size_t)n * 4;
  float px = P[0], py = P[1], pw = P[2], ph = P[3];
  float cx = px + L[0] * 0.1f * pw;
  float cy = py + L[1] * 0.1f * ph;
  float w  = pw * __expf(L[2] * 0.2f);
  float h  = ph * __expf(L[3] * 0.2f);
  float4 bb = make_float4(cx - 0.5f * w, cy - 0.5f * h, cx + 0.5f * w, cy + 0.5f * h);
  *(float4*)(boxes + (size_t)gid * 4) = bb;
}

// ---------------------------------------------------------------------------
// Kernel 2: exact top-200 per (b, class) via MSD radix select, then bitonic
// sort of 256 (200 real + 56 pad) in LDS, then gather boxes.
// ---------------------------------------------------------------------------
__global__ void k_topk(const float* __restrict__ scores_t, // (B,20,N)
                       const float* __restrict__ boxes,    // (B,N,4)
                       float* __restrict__ top_s,          // (B,20,200)
                       float* __restrict__ top_box,        // (B,20,200,4)
                       int N) {
  __shared__ unsigned hist[256];
  __shared__ unsigned s_prefix, s_need, s_cntA, s_cntB;
  __shared__ float sel_s[256];
  __shared__ int   sel_i[256];

  int wg  = blockIdx.x;                 // b*20 + (class-1)
  int tid = threadIdx.x;
  const float* sc = scores_t + (size_t)wg * N;

  if (tid == 0) { s_prefix = 0u; s_need = K_TOP; s_cntA = 0u; s_cntB = 0u; }
  __syncthreads();

  unsigned prefmask = 0u;
  for (int pass = 0; pass < 4; ++pass) {
    int shift = 24 - 8 * pass;
    for (int i = tid; i < 256; i += blockDim.x) hist[i] = 0u;
    __syncthreads();
    unsigned prefix = s_prefix;
    for (int i = tid; i < N; i += blockDim.x) {
      __builtin_prefetch(&sc[i + 4 * blockDim.x], 0, 0);  // global_prefetch_b8
      unsigned k = float_key(sc[i]);
      if ((k & prefmask) == prefix)
        atomicAdd(&hist[(k >> shift) & 255u], 1u);
    }
    __syncthreads();
    if (tid == 0) {
      unsigned need = s_need, cum = 0u; int bin = 0;
      for (int j = 255; j >= 0; --j) {
        if (cum + hist[j] >= need) { bin = j; break; }
        cum += hist[j];
      }
      s_prefix = prefix | ((unsigned)bin << shift);
      s_need   = need - cum;
    }
    prefmask |= (0xFFu << shift);
    __syncthreads();
  }

  unsigned T    = s_prefix;           // exact key of the 200th element
  unsigned need = s_need;             // how many ties (==T) to take
  unsigned nAbove = K_TOP - need;

  for (int i = tid; i < N; i += blockDim.x) {
    unsigned k = float_key(sc[i]);
    if (k > T) {
      unsigned p = atomicAdd(&s_cntA, 1u);
      sel_s[p] = sc[i]; sel_i[p] = i;
    } else if (k == T) {
      unsigned p = atomicAdd(&s_cntB, 1u);
      if (p < need) { sel_s[nAbove + p] = sc[i]; sel_i[nAbove + p] = i; }
    }
  }
  __syncthreads();
  for (int i = K_TOP + tid; i < 256; i += blockDim.x) { sel_s[i] = -INFINITY; sel_i[i] = 0; }
  __syncthreads();

  // bitonic sort, descending, 256 elements, 1 element per thread
  for (int size = 2; size <= 256; size <<= 1) {
    for (int stride = size >> 1; stride > 0; stride >>= 1) {
      int i = tid;
      int j = i ^ stride;
      if (j > i) {
        bool desc = ((i & size) == 0);
        float si = sel_s[i], sj = sel_s[j];
        bool sw = desc ? (si < sj) : (si > sj);
        if (sw) {
          int ii = sel_i[i];
          sel_s[i] = sj; sel_s[j] = si;
          sel_i[i] = sel_i[j]; sel_i[j] = ii;
        }
      }
      __syncthreads();
    }
  }

  if (tid < K_TOP) {
    size_t o = (size_t)wg * K_TOP + tid;
    top_s[o] = sel_s[tid];
    int b = wg / NCM1;
    int idx = sel_i[tid];
    float4 bb = *(const float4*)(boxes + ((size_t)b * N + idx) * 4);
    *(float4*)(top_box + o * 4) = bb;
  }
}

// ---------------------------------------------------------------------------
// Kernel 3: NMS. Suppression bitmasks via WMMA outer-sum union tiles.
// ---------------------------------------------------------------------------
__global__ void k_nms(const float* __restrict__ top_s,
                      const float* __restrict__ top_box,
                      float* __restrict__ out,    // (B,21,200,5)
                      int dummy) {
  __shared__ float sx0[KPAD], sy0[KPAD], sx1[KPAD], sy1[KPAD];
  __shared__ float sarea[KPAD], sscore[KPAD];
  __shared__ unsigned rowmask[KPAD][NWORDS];
  __shared__ unsigned keptmask[NWORDS];

  int wg  = blockIdx.x;               // b*20 + (class-1)
  int b   = wg / NCM1;
  int cc  = wg % NCM1;
  int tid = threadIdx.x;
  size_t base = (size_t)wg * K_TOP;

  for (int i = tid; i < KPAD; i += blockDim.x) {
    float x0 = 0.f, y0 = 0.f, x1 = 0.f, y1 = 0.f, s = 0.f;
    if (i < K_TOP) {
      float4 bb = *(const float4*)(top_box + (base + i) * 4);
      x0 = bb.x; y0 = bb.y; x1 = bb.z; y1 = bb.w;
      s = top_s[base + i];
    }
    sx0[i] = x0; sy0[i] = y0; sx1[i] = x1; sy1[i] = y1;
    sarea[i] = (x1 - x0) * (y1 - y0);
    sscore[i] = s;
#pragma unroll
    for (int w = 0; w < NWORDS; ++w) rowmask[i][w] = 0u;
  }
  __syncthreads();

  int wave = tid >> 5;
  int lane = tid & 31;
  int nwaves = blockDim.x >> 5;

  // 91 upper-triangular 16x16 tiles of the 208x208 pair matrix
  for (int t = wave; t < 91; t += nwaves) {
    int ti = 0, rem = t;
    while (rem >= 13 - ti) { rem -= 13 - ti; ti++; }
    int tj = ti + rem;
    int i0 = ti * 16, j0 = tj * 16;

    // A (16x4): K0 = area_i, K1 = 1, K2 = K3 = 0 (lanes 16..31 hold K2/K3)
    v2f a;
    a.x = (lane < 16) ? sarea[i0 + lane] : 0.f;
    a.y = (lane < 16) ? 1.f : 0.f;
    // B (4x16): row0 = ones, row1 = area_j, rows 2/3 = 0
    v2f bb;
    bb.x = (lane < 16) ? 1.f : 0.f;
    bb.y = (lane < 16) ? sarea[j0 + lane] : 0.f;
    v8f cz = {};
    // D[i][j] = area_i + area_j  -> v_wmma_f32_16x16x4_f32
    v8f d = __builtin_amdgcn_wmma_f32_16x16x4_f32(
        false, a, false, bb, (short)0, cz, false, false);

    int j = j0 + (lane & 15);
    float bx0 = sx0[j], by0 = sy0[j], bx1 = sx1[j], by1 = sy1[j];
    int word = tj >> 1;
    int sh   = (tj & 1) * 16;

#pragma unroll
    for (int r = 0; r < 8; ++r) {
      int i = i0 + r + ((lane >= 16) ? 8 : 0);   // D layout: lanes16-31 -> M=r+8
      float ltx = fmaxf(sx0[i], bx0);
      float lty = fmaxf(sy0[i], by0);
      float rbx = fminf(sx1[i], bx1);
      float rby = fminf(sy1[i], by1);
      float iw = fmaxf(rbx - ltx, 0.f);
      float ih = fmaxf(rby - lty, 0.f);
      float inter = iw * ih;
      float uni = fmaxf(d[r] - inter, 1e-12f);
      // divide-free IoU threshold: inter/uni > t  <=>  inter > t*uni (uni > 0)
      bool cond = inter > NMS_TH * uni;
      unsigned bal = ballot32(cond);
      if (lane == 0) {
        atomicOr(&rowmask[i0 + r][word],     (bal & 0xFFFFu) << sh);
        atomicOr(&rowmask[i0 + r + 8][word], (bal >> 16)     << sh);
      }
    }
  }
  __syncthreads();

  if (tid == 0) {
    unsigned supp[NWORDS] = {0, 0, 0, 0, 0, 0, 0};
    unsigned kept[NWORDS] = {0, 0, 0, 0, 0, 0, 0};
    for (int i = 0; i < K_TOP; ++i) {
      int wi = i >> 5, bi = i & 31;
      bool sup = (supp[wi] >> bi) & 1u;
      if (!sup && sscore[i] > CONF_TH) {
        kept[wi] |= 1u << bi;
        for (int w = wi; w < NWORDS; ++w) {
          unsigned m = rowmask[i][w];
          if (w == wi) m &= (bi == 31) ? 0u : (0xFFFFFFFFu << (bi + 1));  // only j > i
          supp[w] |= m;
        }
      }
    }
#pragma unroll
    for (int w = 0; w < NWORDS; ++w) keptmask[w] = kept[w];
  }
  __syncthreads();

  for (int k = tid; k < K_TOP; k += blockDim.x) {
    bool kp = (keptmask[k >> 5] >> (k & 31)) & 1u;
    size_t o = (((size_t)b * NUM_CLASSES + (cc + 1)) * K_TOP + k) * 5;
    if (kp) {
      out[o + 0] = sscore[k];
      out[o + 1] = sx0[k];
      out[o + 2] = sy0[k];
      out[o + 3] = sx1[k];
      out[o + 4] = sy1[k];
    } else {
      out[o + 0] = 0.f; out[o + 1] = 0.f; out[o + 2] = 0.f;
      out[o + 3] = 0.f; out[o + 4] = 0.f;
    }
  }
}

// ---------------------------------------------------------------------------
// Kernel 4: zero the background-class slab (class 0) of the output.
// ---------------------------------------------------------------------------
__global__ void k_zero_bg(float* __restrict__ out, int B) {
  int per = K_TOP * 5;  // 1000 floats per (b, class) slab
  int gid = blockIdx.x * blockDim.x + threadIdx.x;
  if (gid < B * per) {
    int b = gid / per, r = gid - b * per;
    out[(size_t)b * NUM_CLASSES * per + r] = 0.f;
  }
}

// ---------------------------------------------------------------------------
extern "C" void kernel_launch(void* const* d_in, const int* in_sizes, int n_in,
                              void* d_out, int out_size, void* d_ws, size_t ws_size,
                              hipStream_t stream) {
  const float* loc    = (const float*)d_in[0];   // (B,N,4)
  const float* conf   = (const float*)d_in[1];   // (B,N,21)
  const float* priors = (const float*)d_in[2];   // (N,4)
  float* out = (float*)d_out;                    // (B,21,200,5)

  int N = in_sizes[2] / 4;
  int B = in_sizes[0] / (N * 4);

  // workspace layout (floats)
  size_t scoresN = (size_t)B * NCM1 * N;        // class-major scores
  size_t boxesN  = (size_t)B * N * 4;           // decoded boxes
  size_t topsN   = (size_t)B * NCM1 * K_TOP;    // top scores
  float* scores_t = (float*)d_ws;
  float* boxes    = scores_t + scoresN;
  float* top_s    = boxes + boxesN;
  float* top_box  = top_s + topsN;
  (void)ws_size; (void)n_in; (void)out_size;

  int total = B * N;
  k_softmax_decode<<<(total + 255) / 256, 256, 0, stream>>>(
      loc, conf, priors, scores_t, boxes, B, N);

  k_topk<<<B * NCM1, 256, 0, stream>>>(scores_t, boxes, top_s, top_box, N);

  k_zero_bg<<<(B * K_TOP * 5 + 255) / 256, 256, 0, stream>>>(out, B);

  k_nms<<<B * NCM1, 256, 0, stream>>>(top_s, top_box, out, 0);
}